// SSTInputLayerV2_67997922230596
// MI455X (gfx1250) — compile-verified
//
#include <hip/hip_runtime.h>
#include <cstdint>
#include <cstddef>

// ---------------------------------------------------------------------------
// SSTInputLayerV2 for MI455X (gfx1250, wave32).
// Memory-bound (~660MB HBM traffic => ~28us at 23.3 TB/s). No GEMM content,
// so no WMMA; CDNA5 paths used: 320KB LDS (full 12800-bin histogram per
// block), wave32 shuffle ranking, double-buffered async global->LDS
// (ASYNCcnt) streaming with NT load hint, non-temporal stores to protect the
// 192MB L2, global_prefetch_b8.
// ---------------------------------------------------------------------------

typedef __attribute__((ext_vector_type(4))) float v4f;
typedef __attribute__((ext_vector_type(2))) float v2f;

namespace {
constexpr int kWin        = 12;
constexpr int kNX         = 40;            // ceil(468/12)+1
constexpr int kNY         = 40;
constexpr int kNZ         = 2;             // ceil(1/1)+1
constexpr int kPerSample  = kNX * kNY * kNZ;   // 3200
constexpr int kNumWindows = 4 * kPerSample;    // 12800 (BATCH=4)
constexpr int kChunk      = 4096;
} // namespace

__device__ __forceinline__ unsigned dropTarget(unsigned n) {
  return n < 16u ? 16u : n < 32u ? 32u : n < 64u ? 64u : 144u;
}
__device__ __forceinline__ int dropLevel(unsigned n) {
  return n < 16u ? 0 : n < 32u ? 1 : n < 64u ? 2 : 3;
}
__device__ __forceinline__ int winIndex(int b, int y, int x, int shift) {
  int cx = x + shift, cy = y + shift;           // z-term is always 0
  return b * kPerSample + (cx / kWin) * (kNY * kNZ) + (cy / kWin) * kNZ;
}

// ---- Pass 1: per-chunk LDS histogram of window ids (and win id compute) ----
template <int ROUND>
__global__ void hist_kernel(const int* __restrict__ coords,
                            int* __restrict__ win0, int* __restrict__ win1,
                            const int* __restrict__ keepIn,
                            unsigned* __restrict__ chunkHist, int N) {
  __shared__ unsigned h[kNumWindows];           // 51.2 KB (fits: 320KB/WGP)
  for (int w = threadIdx.x; w < kNumWindows; w += blockDim.x) h[w] = 0u;
  __syncthreads();
  const int c = blockIdx.x;
  const int base = c * kChunk;
  for (int t = threadIdx.x; t < kChunk; t += blockDim.x) {
    int i = base + t;
    if (i < N) {
      if constexpr (ROUND == 0) {
        int4 cd = ((const int4*)coords)[i];     // (b, z, y, x)
        int w0 = winIndex(cd.x, cd.z, cd.w, 12);
        int w1 = winIndex(cd.x, cd.z, cd.w, 6);
        win0[i] = w0;
        win1[i] = w1;
        atomicAdd(&h[w0], 1u);
      } else {
        if (keepIn[i]) atomicAdd(&h[win1[i]], 1u);
      }
    }
  }
  __syncthreads();
  for (int w = threadIdx.x; w < kNumWindows; w += blockDim.x)
    chunkHist[(size_t)c * kNumWindows + w] = h[w];
}

// ---- Pass 2: per-window exclusive scan across chunks; totals -> counts ----
__global__ void scan_kernel(unsigned* __restrict__ chunkHist,
                            unsigned* __restrict__ counts, int numChunks) {
  int w = blockIdx.x * blockDim.x + threadIdx.x;
  if (w >= kNumWindows) return;
  unsigned run = 0;
  for (int c = 0; c < numChunks; ++c) {
    if (c + 1 < numChunks)
      __builtin_prefetch(&chunkHist[(size_t)(c + 1) * kNumWindows + w], 1, 0);
    unsigned t = chunkHist[(size_t)c * kNumWindows + w];
    chunkHist[(size_t)c * kNumWindows + w] = run;   // exclusive base
    run += t;
  }
  counts[w] = run;
}

// ---- Pass 3: stable within-window rank replay (one wave serial per chunk) --
template <int ROUND>
__global__ void rank_kernel(const int* __restrict__ win,
                            const int* __restrict__ keepIn,
                            const unsigned* __restrict__ chunkHist,
                            const unsigned* __restrict__ counts,
                            int* __restrict__ innerOut,
                            int* __restrict__ keepOut, int N) {
  __shared__ unsigned cnt[kNumWindows];
  const int c = blockIdx.x;
  for (int w = threadIdx.x; w < kNumWindows; w += blockDim.x)
    cnt[w] = chunkHist[(size_t)c * kNumWindows + w];
  __syncthreads();
  if (threadIdx.x >= 32) return;                // wave 0 replays in order
  const int lane = threadIdx.x;
  const int base = c * kChunk;
  for (int it = 0; it < kChunk / 32; ++it) {
    int i = base + it * 32 + lane;
    bool valid = i < N;
    int w = valid ? win[i] : 0;
    bool kept = valid;
    if constexpr (ROUND == 1) kept = kept && (keepIn[i] != 0);
    int wk = kept ? w : (-1 - lane);            // unique: matches only itself
    int firstLane = 32, pre = 0, m = 0;
    #pragma unroll
    for (int k = 0; k < 32; ++k) {              // wave32 match/prefix
      int vw = __shfl(wk, k, 32);
      if (vw == wk) {
        if (k < firstLane) firstLane = k;
        if (k < lane) pre++;
        m++;
      }
    }
    unsigned oldv = 0;
    if (kept && lane == firstLane) oldv = atomicAdd(&cnt[w], (unsigned)m);
    oldv = __shfl(oldv, firstLane, 32);
    unsigned rank = oldv + (unsigned)pre;
    if (valid) {
      if constexpr (ROUND == 0) {
        innerOut[i] = (int)rank;
        keepOut[i] = (rank < dropTarget(counts[w])) ? 1 : 0;
      } else {
        innerOut[i] = kept ? (int)rank : -1;
        keepOut[i] = (kept && rank < dropTarget(counts[w])) ? 1 : 0;
      }
    }
  }
}

// ---- Scalar outputs: keep, win0, win1, dl0, dl1, inner0, inner1 -----------
__global__ void scalars_kernel(const int* __restrict__ win0,
                               const int* __restrict__ win1,
                               const int* __restrict__ keep1,
                               const int* __restrict__ keep2,
                               const int* __restrict__ inner0,
                               const int* __restrict__ inner1,
                               const unsigned* __restrict__ counts0,
                               const unsigned* __restrict__ counts1,
                               float* __restrict__ out, int N) {
  int i = blockIdx.x * blockDim.x + threadIdx.x;
  if (i >= N) return;
  int w0 = win0[i], w1 = win1[i];
  float* p = out + (size_t)N * 128;
  p[i] = keep2[i] ? 1.0f : 0.0f;                               // keep
  p[(size_t)N * 1 + i] = (float)w0;                            // win0
  p[(size_t)N * 2 + i] = (float)w1;                            // win1
  p[(size_t)N * 3 + i] = (float)dropLevel(counts0[w0]);        // dl0
  p[(size_t)N * 4 + i] = keep1[i] ? (float)dropLevel(counts1[w1]) : -1.0f;
  p[(size_t)N * 5 + i] = (float)inner0[i];                     // inner0
  p[(size_t)N * 6 + i] = (float)inner1[i];                     // inner1
}

// ---- feat_kept = feat * keep2: double-buffered async global->LDS pipeline --
// Each lane fills and reads back only its own 16B LDS slot, so a wave-level
// s_wait_asynccnt is the only synchronization needed (async loads complete
// in order). One tile is kept in flight to hide HBM latency. Out-of-range
// lanes clamp their address (instead of skipping the issue) so the per-wave
// ASYNCcnt bookkeeping stays deterministic.
__global__ void feat_kernel(const float* __restrict__ feat,
                            const int* __restrict__ keep2,
                            float* __restrict__ out, int nvec) {
  constexpr int kTiles = 4;                     // v4f tiles per block
  __shared__ v4f tile[2][256];                  // 8KB double buffer, LDS off 0
  const int tid = threadIdx.x;
  const int t0 = blockIdx.x * kTiles;           // first tile index

  // Prologue: issue tile 0 into buffer 0.
  {
    int gid = t0 * 256 + tid;
    int cgid = gid < nvec ? gid : (nvec - 1);
    unsigned gOffB = (unsigned)cgid * 16u;
    unsigned ldsOffB = (unsigned)tid * 16u;
    asm volatile("global_load_async_to_lds_b128 %0, %1, %2 th:TH_LOAD_NT"
                 :
                 : "v"(ldsOffB), "v"(gOffB), "s"(feat)
                 : "memory");
  }
  #pragma unroll
  for (int t = 0; t < kTiles; ++t) {
    if (t + 1 < kTiles) {
      // Prefetch next tile into the other buffer, then wait for the OLDER
      // load only (<=1 outstanding): in-order async completion guarantees
      // the buffer we are about to read is resident.
      int gid = (t0 + t + 1) * 256 + tid;
      int cgid = gid < nvec ? gid : (nvec - 1);
      unsigned gOffB = (unsigned)cgid * 16u;
      unsigned ldsOffB = (unsigned)(((t + 1) & 1) * 256 + tid) * 16u;
      asm volatile("global_load_async_to_lds_b128 %0, %1, %2 th:TH_LOAD_NT"
                   :
                   : "v"(ldsOffB), "v"(gOffB), "s"(feat)
                   : "memory");
      asm volatile("s_wait_asynccnt 1" ::: "memory");
    } else {
      asm volatile("s_wait_asynccnt 0" ::: "memory");
    }
    int gid = (t0 + t) * 256 + tid;
    if (gid < nvec) {
      v4f v = tile[t & 1][tid];
      float s = keep2[gid >> 5] ? 1.0f : 0.0f;  // 128 floats = 32 v4f per row
      v4f r = v * s;
      __builtin_nontemporal_store(r, (v4f*)out + gid);  // keep L2 for ws
    }
  }
}

// ---- Sinusoidal position embeddings (both shifts), channel pair per thread
__global__ void pe_kernel(const int* __restrict__ coords,
                          float* __restrict__ out, int N) {
  long t = (long)blockIdx.x * blockDim.x + threadIdx.x;
  if (t >= (long)N * 64) return;
  int i = (int)(t >> 6);
  int k = (int)(t & 63);                        // channel pair: ch = 2k
  int axis = k >> 5;                            // 0 => x, 1 => y
  int f = k & 31;
  int2 yx = *(const int2*)(coords + (size_t)i * 4 + 2);  // (y, x)
  // inv_freq[f] = 10000^(f/32); phase = v / inv_freq = v * 2^(-f*log2(1e4)/32)
  float scale =
      __builtin_amdgcn_exp2f(-(float)f * (13.287712379549449f / 32.0f));
  int ix0 = (yx.y + 12) % 12, iy0 = (yx.x + 12) % 12;
  int ix1 = (yx.y + 6) % 12,  iy1 = (yx.x + 6) % 12;
  float v0 = (axis ? (float)iy0 : (float)ix0) - 6.0f;
  float v1 = (axis ? (float)iy1 : (float)ix1) - 6.0f;
  float p0 = v0 * scale, p1 = v1 * scale;
  float s0, c0, s1, c1;
  __sincosf(p0, &s0, &c0);
  __sincosf(p1, &s1, &c1);
  float* pe0 = out + (size_t)N * 135;           // 128 + 7 scalar arrays
  float* pe1 = out + (size_t)N * 263;
  v2f r0 = {s0, c0}, r1 = {s1, c1};
  __builtin_nontemporal_store(r0, (v2f*)(pe0 + (size_t)i * 128) + k);
  __builtin_nontemporal_store(r1, (v2f*)(pe1 + (size_t)i * 128) + k);
}

extern "C" void kernel_launch(void* const* d_in, const int* in_sizes, int n_in,
                              void* d_out, int out_size, void* d_ws,
                              size_t ws_size, hipStream_t stream) {
  const float* feat = (const float*)d_in[0];
  const int* coords = (const int*)d_in[1];
  const int N = in_sizes[1] / 4;                // 300000
  float* out = (float*)d_out;

  // Workspace layout (all write-before-read each call): ~11.2 MB total.
  int* win0 = (int*)d_ws;
  int* win1 = win0 + N;
  int* keep1 = win1 + N;
  int* keep2 = keep1 + N;
  int* inner0 = keep2 + N;
  int* inner1 = inner0 + N;
  unsigned* counts0 = (unsigned*)(inner1 + N);
  unsigned* counts1 = counts0 + kNumWindows;
  unsigned* chunkHist = counts1 + kNumWindows;  // numChunks * 12800 u32

  const int numChunks = (N + kChunk - 1) / kChunk;   // 74
  dim3 cb(1024), cg(numChunks);
  dim3 sb(256), sg((kNumWindows + 255) / 256);

  // Round 0 (shift = full window, keep starts all-true)
  hist_kernel<0><<<cg, cb, 0, stream>>>(coords, win0, win1, nullptr, chunkHist, N);
  scan_kernel<<<sg, sb, 0, stream>>>(chunkHist, counts0, numChunks);
  rank_kernel<0><<<cg, cb, 0, stream>>>(win0, nullptr, chunkHist, counts0,
                                        inner0, keep1, N);
  // Round 1 (shift = half window, gated by keep1)
  hist_kernel<1><<<cg, cb, 0, stream>>>(coords, win0, win1, keep1, chunkHist, N);
  scan_kernel<<<sg, sb, 0, stream>>>(chunkHist, counts1, numChunks);
  rank_kernel<1><<<cg, cb, 0, stream>>>(win1, keep1, chunkHist, counts1,
                                        inner1, keep2, N);

  scalars_kernel<<<(N + 255) / 256, 256, 0, stream>>>(
      win0, win1, keep1, keep2, inner0, inner1, counts0, counts1, out, N);

  const int nvec = N * 32;                      // N*128 floats as v4f
  const int tilesTotal = (nvec + 255) / 256;
  feat_kernel<<<(tilesTotal + 3) / 4, 256, 0, stream>>>(feat, keep2, out,
                                                        nvec);

  const long peThreads = (long)N * 64;
  pe_kernel<<<(unsigned)((peThreads + 255) / 256), 256, 0, stream>>>(coords,
                                                                     out, N);
}